// NonLinearTPReadOut_3839700763112
// MI455X (gfx1250) — compile-verified
//
#include <hip/hip_runtime.h>
#include <hip/hip_bf16.h>
#include <math.h>

// CDNA5 / gfx1250, wave32.
typedef __attribute__((ext_vector_type(2))) float v2f;
typedef __attribute__((ext_vector_type(8))) float v8f;

#define MULV 128
#define IN_DIM 1152            // 128 * (1 + 3 + 5)
#define OUT_DIM 9
#define WAVES_PER_BLOCK 8
#define ROWS_PER_WAVE 16

__device__ __forceinline__ float silu_f(float x) {
    // x * sigmoid(x)
    return x / (1.0f + __expf(-x));
}

__global__ __launch_bounds__(256) void nltp_kernel(
    const float* __restrict__ x,     // (N, 1152)
    const float* __restrict__ W1,    // (128, 128) row-major [u, v]
    const float* __restrict__ W2,    // (128, 2)   row-major [u, v]
    const float* __restrict__ tpw,   // (3,)
    float* __restrict__ out,         // (N, 9)
    int N)
{
    __shared__ float lds_ab[WAVES_PER_BLOCK][ROWS_PER_WAVE][3][2];

    const int wave = threadIdx.x >> 5;
    const int lane = threadIdx.x & 31;
    const int half = lane >> 4;     // 0: K0/K1 (A) & rows 0-7 (D); 1: K2/K3 & rows 8-15
    const int lm   = lane & 15;     // A: row M; B/D: column N

    const long row0 = ((long)blockIdx.x * WAVES_PER_BLOCK + wave) * ROWS_PER_WAVE;
    if (row0 >= N) return;

    const float INV_SQRT_MUL = 0.08838834764831845f; // 1/sqrt(128)

    // Base of this lane's A-row inside the l=1 block: x[row0+lm, 128 + 3*k + c]
    const float* __restrict__ xr = x + (row0 + lm) * (long)IN_DIM + MULV;

    // Prefetch this lane's 1536-byte l=1 slice (12 cachelines) into WGP$ —
    // it is re-read 3x (once per channel) by strided scalar loads below.
    // Lowers to global_prefetch_b8 on gfx1250.
    #pragma unroll
    for (int p = 0; p < 12; ++p) {
        __builtin_prefetch(xr + p * 32, 0, 0);
    }

    for (int c = 0; c < 3; ++c) {
        // ---- Load A tile (16 x 128) for channel c into ISA 16x16x4 A layout ----
        v2f Areg[32];
        #pragma unroll
        for (int kt = 0; kt < 32; ++kt) {
            const int k = 4 * kt + 2 * half;
            v2f a;
            a.x = xr[3 * k + c];
            a.y = xr[3 * (k + 1) + c];
            Areg[kt] = a;
        }

        float accA[8];  // per-row partial of silu(Y) @ W2[:,0]
        float accB[8];  // per-row partial of silu(Y) @ W2[:,1]
        #pragma unroll
        for (int r = 0; r < 8; ++r) { accA[r] = 0.0f; accB[r] = 0.0f; }

        for (int j = 0; j < 8; ++j) {           // column tiles of W1 (8 x 16 = 128)
            const int ncol = 16 * j + lm;       // this lane's output column
            // Two independent accumulation chains (even/odd K-steps) so the
            // scheduler can overlap dependent WMMA latencies.
            v8f acc0 = {};
            v8f acc1 = {};
            #pragma unroll
            for (int kt = 0; kt < 32; kt += 2) {   // K = 128 in steps of 2x4
                const int k0 = 4 * kt + 2 * half;
                v2f b0;
                b0.x = W1[(long)k0 * MULV + ncol];
                b0.y = W1[(long)(k0 + 1) * MULV + ncol];
                acc0 = __builtin_amdgcn_wmma_f32_16x16x4_f32(
                        false, Areg[kt], false, b0, (short)0, acc0, false, false);

                const int k1 = 4 * (kt + 1) + 2 * half;
                v2f b1;
                b1.x = W1[(long)k1 * MULV + ncol];
                b1.y = W1[(long)(k1 + 1) * MULV + ncol];
                acc1 = __builtin_amdgcn_wmma_f32_16x16x4_f32(
                        false, Areg[kt + 1], false, b1, (short)0, acc1, false, false);
            }
            const float w20 = W2[2 * ncol + 0];
            const float w21 = W2[2 * ncol + 1];
            #pragma unroll
            for (int r = 0; r < 8; ++r) {
                const float yv = (acc0[r] + acc1[r]) * INV_SQRT_MUL;
                const float s  = silu_f(yv);
                accA[r] += s * w20;
                accB[r] += s * w21;
            }
        }

        // ---- Reduce across the 16 columns held by each 16-lane half ----
        #pragma unroll
        for (int r = 0; r < 8; ++r) {
            float va = accA[r];
            float vb = accB[r];
            #pragma unroll
            for (int m = 1; m <= 8; m <<= 1) {
                va += __shfl_xor(va, m, 32);
                vb += __shfl_xor(vb, m, 32);
            }
            if (lm == 0) {
                const int mrow = 8 * half + r;
                lds_ab[wave][mrow][c][0] = va * INV_SQRT_MUL;  // a[n, c]
                lds_ab[wave][mrow][c][1] = vb * INV_SQRT_MUL;  // b[n, c]
            }
        }
    }

    __syncthreads();

    // ---- Tensor-product readout: 16 lanes -> 16 rows, 9 outputs each ----
    if (lane < 16) {
        const int mrow = lane;
        // channel index convention in reference l=2 table: (Y,Z,X) = (0,1,2)
        const float a0 = lds_ab[wave][mrow][0][0];  // a_Y
        const float a1 = lds_ab[wave][mrow][1][0];  // a_Z
        const float a2 = lds_ab[wave][mrow][2][0];  // a_X
        const float b0 = lds_ab[wave][mrow][0][1];  // b_Y
        const float b1 = lds_ab[wave][mrow][1][1];  // b_Z
        const float b2 = lds_ab[wave][mrow][2][1];  // b_X

        const float t0 = tpw[0];
        const float t1 = tpw[1];
        const float t2 = tpw[2];

        const float s2 = 0.7071067811865476f;   // 1/sqrt(2)
        const float s6 = 0.4082482904638631f;   // 1/sqrt(6)
        const float s3 = 0.5773502691896258f;   // 1/sqrt(3)

        float* __restrict__ o = out + (row0 + mrow) * (long)OUT_DIM;

        // out0: tp_w0 * PW0 * (a . b)/sqrt(3)        (C0 = I/sqrt3, PW0 = 1)
        o[0] = t0 * s3 * (a0 * b0 + a1 * b1 + a2 * b2);

        // out1: tp_w1 * sqrt(3) * eps/sqrt(6) -> tp_w1/sqrt(2) * cross(a,b)
        const float f1 = t1 * s2;
        o[1] = f1 * (a1 * b2 - a2 * b1);
        o[2] = f1 * (a2 * b0 - a0 * b2);
        o[3] = f1 * (a0 * b1 - a1 * b0);

        // out2: tp_w2 * sqrt(5) * (M/sqrt(5)) contraction = tp_w2 * M contraction
        o[4] = t2 * s2 * (a2 * b0 + a0 * b2);                       // (X,Y)+(Y,X)
        o[5] = t2 * s2 * (a0 * b1 + a1 * b0);                       // (Y,Z)+(Z,Y)
        o[6] = t2 * s6 * (2.0f * a1 * b1 - a2 * b2 - a0 * b0);      // diag
        o[7] = t2 * s2 * (a2 * b1 + a1 * b2);                       // (X,Z)+(Z,X)
        o[8] = t2 * s2 * (a2 * b2 - a0 * b0);                       // (X,X)-(Y,Y)
    }
}

extern "C" void kernel_launch(void* const* d_in, const int* in_sizes, int n_in,
                              void* d_out, int out_size, void* d_ws, size_t ws_size,
                              hipStream_t stream) {
    const float* x   = (const float*)d_in[0];   // (N, 1152)
    const float* W1  = (const float*)d_in[1];   // (128, 128)
    const float* W2  = (const float*)d_in[2];   // (128, 2)
    const float* tpw = (const float*)d_in[3];   // (3,)
    float* out = (float*)d_out;                 // (N, 9)

    const int N = in_sizes[0] / IN_DIM;
    const int rows_per_block = WAVES_PER_BLOCK * ROWS_PER_WAVE;  // 128
    const int grid = (N + rows_per_block - 1) / rows_per_block;

    nltp_kernel<<<grid, 256, 0, stream>>>(x, W1, W2, tpw, out, N);
}